// GCNEncoder_29721173688328
// MI455X (gfx1250) — compile-verified
//
#include <hip/hip_runtime.h>

typedef __attribute__((ext_vector_type(2))) float v2f;
typedef __attribute__((ext_vector_type(8))) float v8f;

// ---------------------------------------------------------------- utilities
__global__ void fill_kernel(float* __restrict__ p, float v, int n) {
    int i = blockIdx.x * blockDim.x + threadIdx.x;
    if (i < n) p[i] = v;
}

__global__ void deg_kernel(const int* __restrict__ dst, float* __restrict__ deg, int E) {
    int i = blockIdx.x * blockDim.x + threadIdx.x;
    if (i < E) unsafeAtomicAdd(&deg[dst[i]], 1.0f);
}

__global__ void rsqrt_kernel(float* __restrict__ deg, int n) {
    int i = blockIdx.x * blockDim.x + threadIdx.x;
    if (i < n) deg[i] = rsqrtf(deg[i]);   // deg >= 1 always (self-loop)
}

// (v*v)/v via hardware v_rcp_f32: 3 VALU ops, preserves NaN-at-zero.
__device__ __forceinline__ float squash(float v) {
    return (v * v) * __builtin_amdgcn_rcpf(v);
}

// ---------------------------------------------------------------- WMMA GEMM
// H[M,N] = f(X[M,K]) * W[K,N], row-major. One wave per 16x16 tile of H.
// blockDim.x = (N/16)*32 ; gridDim.x = ceil(M/16). Full-precision fp32 WMMA.
template <int K, int N, bool SQUASH>
__global__ __launch_bounds__((N / 16) * 32)
void gemm_wmma(const float* __restrict__ X, const float* __restrict__ W,
               float* __restrict__ H, int M) {
    const int lane  = threadIdx.x & 31;
    const int ntile = threadIdx.x >> 5;     // which 16-col slab of N
    const int half  = lane >> 4;            // 0 or 1
    const int l16   = lane & 15;
    const int row   = blockIdx.x * 16 + l16;         // A row this lane feeds
    const int arow  = row < M ? row : M - 1;          // clamp: keep EXEC full
    const int ncol  = ntile * 16 + l16;               // B/D column this lane feeds
    const float* __restrict__ xrow = X + (long)arow * K;

    v8f c = {0.f, 0.f, 0.f, 0.f, 0.f, 0.f, 0.f, 0.f};

#pragma unroll
    for (int kk = 0; kk < K; kk += 4) {
        const int k0 = kk + half * 2;       // lanes 0-15: K=kk,kk+1 ; 16-31: kk+2,kk+3
        v2f a;
        {
            float2 av = *(const float2*)(xrow + k0);
            if (SQUASH) { av.x = squash(av.x); av.y = squash(av.y); }
            a.x = av.x; a.y = av.y;
        }
        v2f b;
        b.x = W[k0 * N + ncol];
        b.y = W[(k0 + 1) * N + ncol];
        // D = A(16x4,f32) * B(4x16,f32) + C ; exact fp32
        c = __builtin_amdgcn_wmma_f32_16x16x4_f32(
                /*neg_a=*/false, a, /*neg_b=*/false, b,
                /*c_mod=*/(short)0, c, /*reuse_a=*/false, /*reuse_b=*/false);
    }

    // C/D layout: VGPR i -> row (i + 8*half), col = l16 (within tile)
#pragma unroll
    for (int i = 0; i < 8; ++i) {
        int r = blockIdx.x * 16 + i + half * 8;
        if (r < M) H[(long)r * N + ncol] = c[i];
    }
}

// ---------------------------------------------------------------- aggregation
// One wave32 per edge: acc[dst, :] += dinv[src]*dinv[dst] * h[src, :]
template <int C>
__global__ void scatter_kernel(const float* __restrict__ h, const int* __restrict__ src,
                               const int* __restrict__ dst, const float* __restrict__ dinv,
                               float* __restrict__ acc, int E) {
    int gid  = blockIdx.x * blockDim.x + threadIdx.x;
    int e    = gid >> 5;
    int lane = gid & 31;
    if (e >= E) return;
    int s = src[e], d = dst[e];
    float p = dinv[s] * dinv[d];
    if (C == 64) {
        float2 v = *(const float2*)(h + (long)s * 64 + lane * 2);
        unsafeAtomicAdd(&acc[(long)d * 64 + lane * 2 + 0], p * v.x);
        unsafeAtomicAdd(&acc[(long)d * 64 + lane * 2 + 1], p * v.y);
    } else {  // C == 32
        float v = h[(long)s * 32 + lane];
        unsafeAtomicAdd(&acc[(long)d * 32 + lane], p * v);
    }
}

// out = (acc + dinv[i]^2 * h + bias) , optional ReLU  (self-loop folded in)
template <int C, bool RELU>
__global__ void finalize_kernel(const float* __restrict__ acc, const float* __restrict__ h,
                                const float* __restrict__ dinv, const float* __restrict__ bias,
                                float* __restrict__ out, int Nn) {
    int idx = blockIdx.x * blockDim.x + threadIdx.x;
    if (idx >= Nn * C) return;
    int i = idx / C, ch = idx & (C - 1);
    float di = dinv[i];
    float v = acc[idx] + di * di * h[idx] + bias[ch];
    out[idx] = RELU ? fmaxf(v, 0.0f) : v;
}

// ---------------------------------------------------------------- launcher
static inline size_t align256(size_t n) { return (n + 255) & ~size_t(255); }

extern "C" void kernel_launch(void* const* d_in, const int* in_sizes, int n_in,
                              void* d_out, int out_size, void* d_ws, size_t ws_size,
                              hipStream_t stream) {
    const float* x  = (const float*)d_in[0];
    const int*   ei = (const int*)d_in[1];
    const float* W1 = (const float*)d_in[2];
    const float* b1 = (const float*)d_in[3];
    const float* W2 = (const float*)d_in[4];
    const float* b2 = (const float*)d_in[5];
    float* out = (float*)d_out;

    const int hid  = in_sizes[3];          // 64
    const int outc = in_sizes[5];          // 32
    const int inc  = in_sizes[2] / hid;    // 128
    const int Nn   = in_sizes[0] / inc;    // 100000
    const int E    = in_sizes[1] / 2;      // 1600000
    const int* src = ei;
    const int* dstv = ei + E;
    (void)outc;

    char* ws = (char*)d_ws;
    float* deg = (float*)ws;  ws += align256((size_t)Nn * 4);          // deg -> dinv (in place)
    float* h1  = (float*)ws;  ws += (size_t)Nn * 64 * 4;               // X W1
    float* a1  = (float*)ws;  ws += (size_t)Nn * 64 * 4;               // layer-1 accum / relu out
    float* h2  = (float*)ws;                                           // a1 W2

    const int T = 256;
    // init: deg=1 (self-loop), accumulators = 0
    fill_kernel<<<(Nn + T - 1) / T, T, 0, stream>>>(deg, 1.0f, Nn);
    fill_kernel<<<(Nn * 64 + T - 1) / T, T, 0, stream>>>(a1, 0.0f, Nn * 64);
    fill_kernel<<<(Nn * 32 + T - 1) / T, T, 0, stream>>>(out, 0.0f, Nn * 32);

    // degrees and dinv = rsqrt(deg)
    deg_kernel<<<(E + T - 1) / T, T, 0, stream>>>(dstv, deg, E);
    rsqrt_kernel<<<(Nn + T - 1) / T, T, 0, stream>>>(deg, Nn);

    // layer 1: GEMM (fp32 WMMA, (v*v)/v fused) -> scatter -> bias+selfloop+relu
    gemm_wmma<128, 64, true><<<(Nn + 15) / 16, 128, 0, stream>>>(x, W1, h1, Nn);
    scatter_kernel<64><<<(E + 7) / 8, T, 0, stream>>>(h1, src, dstv, deg, a1, E);
    finalize_kernel<64, true><<<(Nn * 64 + T - 1) / T, T, 0, stream>>>(a1, h1, deg, b1, a1, Nn);

    // layer 2: GEMM -> scatter into d_out -> bias+selfloop
    gemm_wmma<64, 32, false><<<(Nn + 15) / 16, 64, 0, stream>>>(a1, W2, h2, Nn);
    scatter_kernel<32><<<(E + 7) / 8, T, 0, stream>>>(h2, src, dstv, deg, out, E);
    finalize_kernel<32, false><<<(Nn * 32 + T - 1) / T, T, 0, stream>>>(out, h2, deg, b2, out, Nn);
}